// SpatialTransformerNetwork_70866960384431
// MI455X (gfx1250) — compile-verified
//
#include <hip/hip_runtime.h>

typedef float v2f __attribute__((ext_vector_type(2)));
typedef float v8f __attribute__((ext_vector_type(8)));

#define IMG_PIX   784          // 28*28
#define IMG_W     28
#define IMG_H     28
#define IMG_ELEMS 2352         // 28*28*3

#define C1_OC    8
#define C1_K     147           // 3*7*7
#define C1_KP    148           // padded to multiple of 4
#define C1_OW    22
#define C1_M     484           // 22*22
#define C1_TILES 31            // ceil(484/16)

#define P1_W     11
#define P1_PIX   121

#define C2_OC    10
#define C2_K     200           // 8*5*5
#define C2_OW    7
#define C2_M     49            // 7*7
#define C2_TILES 4             // ceil(49/16)

__global__ __launch_bounds__(256) void stn_fused_kernel(
    const float* __restrict__ x,
    const float* __restrict__ w1, const float* __restrict__ b1,
    const float* __restrict__ w2, const float* __restrict__ b2,
    const float* __restrict__ fw1, const float* __restrict__ fb1,
    const float* __restrict__ fw2, const float* __restrict__ fb2,
    float* __restrict__ out)
{
    __shared__ float s_img[3 * IMG_PIX];     // image CHW
    __shared__ float s_w1 [16 * C1_KP];      // conv1 weights, zero padded
    __shared__ float s_c1 [C1_OC * C1_M];    // conv1 raw output [8][484]
    __shared__ float s_p1 [C1_OC * P1_PIX];  // pooled+relu [8][11][11]
    __shared__ float s_w2 [16 * C2_K];       // conv2 weights, zero padded
    __shared__ float s_c2 [C2_OC * C2_M];    // conv2 raw output [10][49]
    __shared__ float s_h  [96];              // flattened pooled conv2 (90 used)
    __shared__ float s_h1 [32];              // fc1 output
    __shared__ float s_theta[6];
    __shared__ int   s_ko1[C1_KP];           // im2col k -> lds offset (conv1)
    __shared__ int   s_ko2[C2_K];            // im2col k -> lds offset (conv2)

    const int tid = threadIdx.x;
    const int img = blockIdx.x;
    const long base = (long)img * IMG_ELEMS;

    // ---- stage 0: load image HWC -> CHW, weights -> padded LDS, k-offset tables
    for (int j = tid; j < IMG_ELEMS; j += 256) {
        int c = j / IMG_PIX;
        int p = j - c * IMG_PIX;
        s_img[j] = x[base + (long)p * 3 + c];
    }
    for (int j = tid; j < 16 * C1_KP; j += 256) {
        int ch = j / C1_KP, k = j - ch * C1_KP;
        s_w1[j] = (ch < C1_OC && k < C1_K) ? w1[ch * C1_K + k] : 0.0f;
    }
    for (int j = tid; j < 16 * C2_K; j += 256) {
        int ch = j / C2_K, k = j - ch * C2_K;
        s_w2[j] = (ch < C2_OC) ? w2[ch * C2_K + k] : 0.0f;
    }
    for (int k = tid; k < C1_KP; k += 256) {
        if (k < C1_K) {
            int c = k / 49, r = (k % 49) / 7, s = k % 7;
            s_ko1[k] = c * IMG_PIX + r * IMG_W + s;
        } else {
            s_ko1[k] = 0;                        // weight is 0 there -> harmless
        }
    }
    for (int k = tid; k < C2_K; k += 256) {
        int c = k / 25, r = (k % 25) / 5, s = k % 5;
        s_ko2[k] = c * P1_PIX + r * P1_W + s;
    }
    __syncthreads();

    const int lane = tid & 31;
    const int wave = tid >> 5;        // 8 waves
    const int half = lane >> 4;       // 0: K+0/1 | 1: K+2/3 (A); rows K / K+2 (B)
    const int mrow = lane & 15;       // A-fragment row within tile
    const int nch  = lane & 15;       // B-fragment / C-fragment column

    // ---- stage 1: conv1 as im2col GEMM, exact fp32 WMMA 16x16x4
    {
        const float bias = (nch < C1_OC) ? b1[nch] : 0.0f;
        for (int t = wave; t < C1_TILES; t += 8) {           // wave-uniform loop
            int m  = t * 16 + mrow;
            int mc = (m < C1_M) ? m : (C1_M - 1);            // clamp pad rows
            int abase = (mc / C1_OW) * IMG_W + (mc % C1_OW);
            v8f acc = {};
            for (int kk = 0; kk < C1_KP; kk += 4) {
                int kb = kk + half * 2;
                v2f a, b;
                a.x = s_img[s_ko1[kb]     + abase];
                a.y = s_img[s_ko1[kb + 1] + abase];
                b.x = s_w1[nch * C1_KP + kb];
                b.y = s_w1[nch * C1_KP + kb + 1];
                acc = __builtin_amdgcn_wmma_f32_16x16x4_f32(
                        false, a, false, b, (short)0, acc, false, false);
            }
            if (nch < C1_OC) {
                #pragma unroll
                for (int r = 0; r < 8; ++r) {
                    int M = t * 16 + r + half * 8;
                    if (M < C1_M) s_c1[nch * C1_M + M] = acc[r] + bias;
                }
            }
        }
    }
    __syncthreads();

    // ---- stage 2: maxpool 2x2 stride 2 + relu -> [8][11][11]
    for (int j = tid; j < C1_OC * P1_PIX; j += 256) {
        int ch = j / P1_PIX;
        int p  = j - ch * P1_PIX;
        int py = p / P1_W, px = p - py * P1_W;
        const float* rp = &s_c1[ch * C1_M + (2 * py) * C1_OW + 2 * px];
        float v = fmaxf(fmaxf(rp[0], rp[1]), fmaxf(rp[C1_OW], rp[C1_OW + 1]));
        s_p1[j] = fmaxf(v, 0.0f);
    }
    __syncthreads();

    // ---- stage 3: conv2 as im2col GEMM (M=49, K=200), fp32 WMMA
    {
        const float bias = (nch < C2_OC) ? b2[nch] : 0.0f;
        if (wave < C2_TILES) {                               // wave-uniform branch
            int t  = wave;
            int m  = t * 16 + mrow;
            int mc = (m < C2_M) ? m : (C2_M - 1);
            int abase = (mc / C2_OW) * P1_W + (mc % C2_OW);
            v8f acc = {};
            for (int kk = 0; kk < C2_K; kk += 4) {
                int kb = kk + half * 2;
                v2f a, b;
                a.x = s_p1[s_ko2[kb]     + abase];
                a.y = s_p1[s_ko2[kb + 1] + abase];
                b.x = s_w2[nch * C2_K + kb];
                b.y = s_w2[nch * C2_K + kb + 1];
                acc = __builtin_amdgcn_wmma_f32_16x16x4_f32(
                        false, a, false, b, (short)0, acc, false, false);
            }
            if (nch < C2_OC) {
                #pragma unroll
                for (int r = 0; r < 8; ++r) {
                    int M = t * 16 + r + half * 8;
                    if (M < C2_M) s_c2[nch * C2_M + M] = acc[r] + bias;
                }
            }
        }
    }
    __syncthreads();

    // ---- stage 4: maxpool 7->3 + relu, flatten to 90 (ch*9 + py*3 + px)
    for (int j = tid; j < 90; j += 256) {
        int ch = j / 9;
        int p  = j - ch * 9;
        int py = p / 3, px = p - py * 3;
        const float* rp = &s_c2[ch * C2_M + (2 * py) * C2_OW + 2 * px];
        float v = fmaxf(fmaxf(rp[0], rp[1]), fmaxf(rp[C2_OW], rp[C2_OW + 1]));
        s_h[j] = fmaxf(v, 0.0f);
    }
    __syncthreads();

    // ---- stage 5: fc1 (32x90) + relu ; fc2 (6x32) -> theta
    if (tid < 32) {
        float acc = fb1[tid];
        for (int j = 0; j < 90; ++j) acc += fw1[tid * 90 + j] * s_h[j];
        s_h1[tid] = fmaxf(acc, 0.0f);
    }
    __syncthreads();
    if (tid < 6) {
        float acc = fb2[tid];
        for (int j = 0; j < 32; ++j) acc += fw2[tid * 32 + j] * s_h1[j];
        s_theta[tid] = acc;
    }
    __syncthreads();

    // ---- stage 6: affine grid + bilinear sample (zeros pad, align_corners=False)
    const float t00 = s_theta[0], t01 = s_theta[1], t02 = s_theta[2];
    const float t10 = s_theta[3], t11 = s_theta[4], t12 = s_theta[5];
    for (int j = tid; j < IMG_ELEMS; j += 256) {
        int y   = j / (IMG_W * 3);
        int rem = j - y * IMG_W * 3;
        int xq  = rem / 3;
        int c   = rem - xq * 3;
        float xs = (2.0f * xq + 1.0f) * (1.0f / IMG_W) - 1.0f;
        float ys = (2.0f * y  + 1.0f) * (1.0f / IMG_H) - 1.0f;
        float gx = t00 * xs + t01 * ys + t02;
        float gy = t10 * xs + t11 * ys + t12;
        float ix = ((gx + 1.0f) * IMG_W - 1.0f) * 0.5f;
        float iy = ((gy + 1.0f) * IMG_H - 1.0f) * 0.5f;
        float x0f = floorf(ix), y0f = floorf(iy);
        float wx1 = ix - x0f, wy1 = iy - y0f;
        float wx0 = 1.0f - wx1, wy0 = 1.0f - wy1;
        int x0 = (int)x0f, y0 = (int)y0f;
        int x1 = x0 + 1,  y1 = y0 + 1;
        const float* ic = &s_img[c * IMG_PIX];
        bool bx0 = (x0 >= 0) & (x0 < IMG_W);
        bool bx1 = (x1 >= 0) & (x1 < IMG_W);
        bool by0 = (y0 >= 0) & (y0 < IMG_H);
        bool by1 = (y1 >= 0) & (y1 < IMG_H);
        float v00 = (bx0 & by0) ? ic[y0 * IMG_W + x0] : 0.0f;
        float v10 = (bx1 & by0) ? ic[y0 * IMG_W + x1] : 0.0f;
        float v01 = (bx0 & by1) ? ic[y1 * IMG_W + x0] : 0.0f;
        float v11 = (bx1 & by1) ? ic[y1 * IMG_W + x1] : 0.0f;
        out[base + j] = v00 * wx0 * wy0 + v10 * wx1 * wy0
                      + v01 * wx0 * wy1 + v11 * wx1 * wy1;
    }
}

extern "C" void kernel_launch(void* const* d_in, const int* in_sizes, int n_in,
                              void* d_out, int out_size, void* d_ws, size_t ws_size,
                              hipStream_t stream) {
    const float* x   = (const float*)d_in[0];
    const float* w1  = (const float*)d_in[1];
    const float* b1  = (const float*)d_in[2];
    const float* w2  = (const float*)d_in[3];
    const float* b2  = (const float*)d_in[4];
    const float* fw1 = (const float*)d_in[5];
    const float* fb1 = (const float*)d_in[6];
    const float* fw2 = (const float*)d_in[7];
    const float* fb2 = (const float*)d_in[8];
    float* out = (float*)d_out;

    int n_img = in_sizes[0] / IMG_ELEMS;   // 64*256 = 16384 images
    stn_fused_kernel<<<n_img, 256, 0, stream>>>(x, w1, b1, w2, b2,
                                                fw1, fb1, fw2, fb2, out);
}